// _LSTMEncoderDecoder_26388279066959
// MI455X (gfx1250) — compile-verified
//
#include <hip/hip_runtime.h>
#include <math.h>

// Problem dims (fixed by the reference)
#define AA 64
#define BB 256
#define DD 128
#define HH 512

#define DEC_WGS   64   // decoder workgroups (one per WGP, weights LDS-resident)
#define DEC_THR   128  // threads per decoder WG
#define ROWS_WG   32   // 4 gates x 8 hidden indices per WG
#define ROW_LEN   640  // concat [Wih row (128) | Whh row (512)]

typedef __attribute__((ext_vector_type(2))) float v2f;
typedef __attribute__((ext_vector_type(8))) float v8f;
typedef __attribute__((ext_vector_type(4))) unsigned int v4u;
typedef __attribute__((ext_vector_type(4))) int v4i_t;
typedef __attribute__((ext_vector_type(8))) int v8i_t;

__device__ __forceinline__ float sigmoidf_(float x) { return 1.0f / (1.0f + __expf(-x)); }

__global__ void zero_kernel(float* __restrict__ p, int n) {
    int i = blockIdx.x * blockDim.x + threadIdx.x;
    if (i < n) p[i] = 0.0f;
}

// -------------------------------------------------------------------------
// Encoder step (unchanged from round 1): one wave32 per 16x16 tile of
// (B=256,H=512), all 4 gates accumulated with V_WMMA_F32_16X16X4_F32.
// -------------------------------------------------------------------------
__global__ __launch_bounds__(128) void enc_step_kernel(
    const float* __restrict__ x,    // (BB, DD)
    const float* __restrict__ Wih,  // (4H, DD)
    const float* __restrict__ Whh,  // (4H, HH)
    const float* __restrict__ bih,
    const float* __restrict__ bhh,
    const float* __restrict__ h_in,
    const float* __restrict__ c_in,
    float* __restrict__ h_out,
    float* __restrict__ c_out)
{
    const int lane  = threadIdx.x & 31;
    const int wave  = threadIdx.x >> 5;
    const int tile  = blockIdx.x * 4 + wave;
    const int m0    = (tile & 15) * 16;
    const int n0    = (tile >> 4) * 16;
    const int lhalf = lane >> 4;
    const int l15   = lane & 15;

    v8f acc0 = {}, acc1 = {}, acc2 = {}, acc3 = {};

    {
        const float* arow = x + (size_t)(m0 + l15) * DD;
        const float* b0 = Wih + (size_t)(0 * HH + n0 + l15) * DD;
        const float* b1 = Wih + (size_t)(1 * HH + n0 + l15) * DD;
        const float* b2 = Wih + (size_t)(2 * HH + n0 + l15) * DD;
        const float* b3 = Wih + (size_t)(3 * HH + n0 + l15) * DD;
        for (int k = 0; k < DD; k += 4) {
            const int kk = k + 2 * lhalf;
            v2f a; a.x = arow[kk]; a.y = arow[kk + 1];
            v2f b;
            b.x = b0[kk]; b.y = b0[kk + 1];
            acc0 = __builtin_amdgcn_wmma_f32_16x16x4_f32(false, a, false, b, (short)0, acc0, false, false);
            b.x = b1[kk]; b.y = b1[kk + 1];
            acc1 = __builtin_amdgcn_wmma_f32_16x16x4_f32(false, a, false, b, (short)0, acc1, false, false);
            b.x = b2[kk]; b.y = b2[kk + 1];
            acc2 = __builtin_amdgcn_wmma_f32_16x16x4_f32(false, a, false, b, (short)0, acc2, false, false);
            b.x = b3[kk]; b.y = b3[kk + 1];
            acc3 = __builtin_amdgcn_wmma_f32_16x16x4_f32(false, a, false, b, (short)0, acc3, false, false);
        }
    }
    {
        const float* arow = h_in + (size_t)(m0 + l15) * HH;
        const float* b0 = Whh + (size_t)(0 * HH + n0 + l15) * HH;
        const float* b1 = Whh + (size_t)(1 * HH + n0 + l15) * HH;
        const float* b2 = Whh + (size_t)(2 * HH + n0 + l15) * HH;
        const float* b3 = Whh + (size_t)(3 * HH + n0 + l15) * HH;
        for (int k = 0; k < HH; k += 4) {
            const int kk = k + 2 * lhalf;
            v2f a; a.x = arow[kk]; a.y = arow[kk + 1];
            v2f b;
            b.x = b0[kk]; b.y = b0[kk + 1];
            acc0 = __builtin_amdgcn_wmma_f32_16x16x4_f32(false, a, false, b, (short)0, acc0, false, false);
            b.x = b1[kk]; b.y = b1[kk + 1];
            acc1 = __builtin_amdgcn_wmma_f32_16x16x4_f32(false, a, false, b, (short)0, acc1, false, false);
            b.x = b2[kk]; b.y = b2[kk + 1];
            acc2 = __builtin_amdgcn_wmma_f32_16x16x4_f32(false, a, false, b, (short)0, acc2, false, false);
            b.x = b3[kk]; b.y = b3[kk + 1];
            acc3 = __builtin_amdgcn_wmma_f32_16x16x4_f32(false, a, false, b, (short)0, acc3, false, false);
        }
    }

    const int n = n0 + l15;
    const float bi = bih[0 * HH + n] + bhh[0 * HH + n];
    const float bf = bih[1 * HH + n] + bhh[1 * HH + n];
    const float bg = bih[2 * HH + n] + bhh[2 * HH + n];
    const float bo = bih[3 * HH + n] + bhh[3 * HH + n];
#pragma unroll
    for (int r = 0; r < 8; ++r) {
        const int m = m0 + r + lhalf * 8;
        const float gi = acc0[r] + bi;
        const float gf = acc1[r] + bf;
        const float gg = acc2[r] + bg;
        const float go = acc3[r] + bo;
        const float c_old = c_in[(size_t)m * HH + n];
        const float c_new = sigmoidf_(gf) * c_old + sigmoidf_(gi) * tanhf(gg);
        const float h_new = sigmoidf_(go) * tanhf(c_new);
        c_out[(size_t)m * HH + n] = c_new;
        h_out[(size_t)m * HH + n] = h_new;
    }
}

// -------------------------------------------------------------------------
// TDM: async copy of one 128-float x-vector from global into LDS.
// D# per CDNA5 ISA ch.8: group0 = {count|flags, lds_addr, global_addr, type=2},
// group1 = dims/strides (data_size=4B, tensor_dim0=tile_dim0=128, 1 row).
// Issued by wave 0 only (EXEC is ignored by TDM; one issue per wave).
// -------------------------------------------------------------------------
__device__ __forceinline__ void tdm_load_x(const float* gsrc, unsigned lds_off) {
    const unsigned long long ga = (unsigned long long)(size_t)(const void*)gsrc;
    v4u g0;
    g0.x = 1u;                                                  // count=1
    g0.y = lds_off;                                             // LDS byte addr
    g0.z = (unsigned)ga;                                        // global_addr lo
    g0.w = (unsigned)((ga >> 32) & 0x1FFFFFFu) | (2u << 30);    // addr hi | type=2
    v8i_t g1;
    g1[0] = 0x20000;      // workgroup_mask=0, data_size=2 (4 bytes)
    g1[1] = (128 << 16);  // tensor_dim0 lo16 = 128
    g1[2] = (1 << 16);    // tensor_dim0 hi | tensor_dim1 lo16 = 1
    g1[3] = (128 << 16);  // tensor_dim1 hi | tile_dim0 = 128
    g1[4] = 1;            // tile_dim1 = 1, tile_dim2 = 0
    g1[5] = 128;          // tensor_dim0_stride lo32
    g1[6] = (128 << 16);  // stride0 hi16 | tensor_dim1_stride lo16
    g1[7] = 0;
    v4i_t z4 = {0, 0, 0, 0};
#if __has_include(<hip/amd_detail/amd_gfx1250_TDM.h>)
    v8i_t z8 = {0, 0, 0, 0, 0, 0, 0, 0};
    __builtin_amdgcn_tensor_load_to_lds(g0, g1, z4, z4, z8, 0);   // therock 6-arg form
#else
    __builtin_amdgcn_tensor_load_to_lds(g0, g1, z4, z4, 0);       // ROCm 7.2 5-arg form
#endif
}

// -------------------------------------------------------------------------
// Decoder: 16384-step serial LSTM chain distributed over 64 WGPs.
//  - Each WG keeps its 32 weight rows (80 KB) resident in LDS for the
//    whole kernel: zero per-step weight traffic.
//  - Per step: 40 float4 FMAs/thread, LDS reduction, local h/c update,
//    h exchanged via L2 (agent-scope atomics) + 1 global barrier/step
//    (double-buffered hbuf makes one barrier sufficient).
//  - Next step's x vector prefetched via the Tensor Data Mover into an
//    LDS double buffer (s_wait_tensorcnt before consumption).
// -------------------------------------------------------------------------
__global__ __launch_bounds__(DEC_THR) void dec_kernel(
    const float* __restrict__ input, // (AA, BB, DD)
    const float* __restrict__ Wih,   // (4H, DD)
    const float* __restrict__ Whh,   // (4H, HH)
    const float* __restrict__ bih,
    const float* __restrict__ bhh,
    const float* __restrict__ linW,  // (DD, HH)
    const float* __restrict__ linb,  // (DD)
    const float* __restrict__ hN,    // (BB, HH)
    const float* __restrict__ cN,    // (BB, HH)
    float* __restrict__ hbuf,        // (2, HH) h exchange, double buffered
    unsigned* __restrict__ bar,      // global barrier counter (zeroed)
    float* __restrict__ out)         // (AA, BB, DD)
{
    __shared__ __align__(16) float sW[ROWS_WG * ROW_LEN]; // 80 KB weight slice
    __shared__ __align__(16) float sx[2][DD];             // x double buffer (TDM dest)
    __shared__ __align__(16) float shh[HH];               // full h vector
    __shared__ __align__(16) float sbias[ROWS_WG];
    __shared__ __align__(16) float spart[DEC_THR];
    __shared__ __align__(16) float sgate[ROWS_WG];
    __shared__ __align__(16) float sc[8];                 // this WG's c slice

    const int tid = threadIdx.x;
    const int wg  = blockIdx.x;
    const int n0  = wg * 8;          // hidden indices [n0, n0+8)
    const int row = tid & (ROWS_WG - 1);
    const int p   = tid >> 5;        // 4 k-partitions per row

    // ---- one-time preload: weight slice + bias into LDS ----
    for (int idx = tid; idx < ROWS_WG * ROW_LEN; idx += DEC_THR) {
        const int r = idx / ROW_LEN;
        const int k = idx - r * ROW_LEN;
        const int grow = (r >> 3) * HH + n0 + (r & 7);
        sW[idx] = (k < DD) ? Wih[(size_t)grow * DD + k]
                           : Whh[(size_t)grow * HH + (k - DD)];
    }
    for (int r = tid; r < ROWS_WG; r += DEC_THR) {
        const int grow = (r >> 3) * HH + n0 + (r & 7);
        sbias[r] = bih[grow] + bhh[grow];
    }
    for (int k = tid; k < HH; k += DEC_THR)
        shh[k] = hN[(size_t)(BB - 1) * HH + k];
    if (tid < 8)
        sc[tid] = cN[(size_t)(BB - 1) * HH + n0 + tid];
    __syncthreads();

    // prologue: prefetch x for global step 0 (col b=255, t=0)
    if (tid < 32)
        tdm_load_x(input + (size_t)(BB - 1) * DD, (unsigned)(size_t)(void*)&sx[0][0]);

    for (int gs = 0; gs < BB * AA; ++gs) {
        const int i     = gs >> 6;
        const int t     = gs & (AA - 1);
        const int b_idx = BB - 1 - i;
        const int cur   = gs & 1;

        // out_i = h @ linW^T + linb, computed at column start from current h
        if (t == 0 && tid < 2) {
            const int d = wg * 2 + tid;
            float acc = linb[d];
            const float4* w4 = (const float4*)(linW + (size_t)d * HH);
            const float4* h4 = (const float4*)shh;
#pragma unroll 8
            for (int k = 0; k < HH / 4; ++k) {
                const float4 w = w4[k], v = h4[k];
                acc += w.x * v.x + w.y * v.y + w.z * v.z + w.w * v.w;
            }
            for (int a = 0; a < AA; ++a)
                out[((size_t)a * BB + b_idx) * DD + d] = acc;
        }

        // x[cur] ready (TDM issued last iteration / prologue)
        if (tid < 32) __builtin_amdgcn_s_wait_tensorcnt(0);
        __syncthreads();

        // prefetch next step's x into the other buffer
        if (tid < 32 && gs + 1 < BB * AA) {
            const int gn  = gs + 1;
            const int bn  = BB - 1 - (gn >> 6);
            const int tn  = gn & (AA - 1);
            tdm_load_x(input + ((size_t)tn * BB + bn) * DD,
                       (unsigned)(size_t)(void*)&sx[cur ^ 1][0]);
        }

        // ---- gate partials: each thread 40 float4 FMAs over [x|h] ----
        {
            const float* wr = &sW[row * ROW_LEN];
            const float* xc = &sx[cur][0];
            float acc = 0.0f;
            const int k0 = p * 160, k1 = k0 + 160;
            const int xe = (k1 < DD) ? k1 : DD;
            for (int k = k0; k < xe; k += 4) {
                const float4 w = *(const float4*)&wr[k];
                const float4 v = *(const float4*)&xc[k];
                acc += w.x * v.x + w.y * v.y + w.z * v.z + w.w * v.w;
            }
            const int hs = (k0 > DD) ? k0 : DD;
            for (int k = hs; k < k1; k += 4) {
                const float4 w = *(const float4*)&wr[k];
                const float4 v = *(const float4*)&shh[k - DD];
                acc += w.x * v.x + w.y * v.y + w.z * v.z + w.w * v.w;
            }
            spart[tid] = acc;
        }
        __syncthreads();

        if (tid < ROWS_WG)
            sgate[tid] = sbias[tid] + spart[tid] + spart[tid + 32]
                       + spart[tid + 64] + spart[tid + 96];
        __syncthreads();

        // ---- local h/c update for this WG's 8 hidden indices ----
        if (tid < 8) {
            const float gi = sgate[tid];
            const float gf = sgate[8 + tid];
            const float gg = sgate[16 + tid];
            const float go = sgate[24 + tid];
            const float c_new = sigmoidf_(gf) * sc[tid] + sigmoidf_(gi) * tanhf(gg);
            sc[tid] = c_new;
            const float h_new = sigmoidf_(go) * tanhf(c_new);
            __hip_atomic_store(&hbuf[(size_t)cur * HH + n0 + tid], h_new,
                               __ATOMIC_RELAXED, __HIP_MEMORY_SCOPE_AGENT);
            __threadfence();
        }
        __syncthreads();

        // ---- one global barrier per step (release/acquire, monotonic) ----
        if (tid == 0) {
            __hip_atomic_fetch_add(bar, 1u, __ATOMIC_RELEASE, __HIP_MEMORY_SCOPE_AGENT);
            const unsigned target = (unsigned)(gs + 1) * DEC_WGS;
            while (__hip_atomic_load(bar, __ATOMIC_ACQUIRE, __HIP_MEMORY_SCOPE_AGENT) < target)
                __builtin_amdgcn_s_sleep(1);
        }
        __syncthreads();

        // ---- gather the full updated h (agent-scope, bypass stale WGP$) ----
        for (int k = tid; k < HH; k += DEC_THR)
            shh[k] = __hip_atomic_load(&hbuf[(size_t)cur * HH + k],
                                       __ATOMIC_RELAXED, __HIP_MEMORY_SCOPE_AGENT);
        __syncthreads();
    }
}

// -------------------------------------------------------------------------
// Host orchestration. d_ws layout (floats):
//   [0,BH) h ping | [BH,2BH) c ping | [2BH,3BH) h pong | [3BH,4BH) c pong
//   [4BH, 4BH+1024) decoder h exchange (2 x 512) | [4BH+1024] barrier u32
// ~2 MB + 4 KB scratch.
// -------------------------------------------------------------------------
extern "C" void kernel_launch(void* const* d_in, const int* in_sizes, int n_in,
                              void* d_out, int out_size, void* d_ws, size_t ws_size,
                              hipStream_t stream) {
    const float* input = (const float*)d_in[0];
    const float* eWih  = (const float*)d_in[1];
    const float* eWhh  = (const float*)d_in[2];
    const float* ebih  = (const float*)d_in[3];
    const float* ebhh  = (const float*)d_in[4];
    const float* dWih  = (const float*)d_in[5];
    const float* dWhh  = (const float*)d_in[6];
    const float* dbih  = (const float*)d_in[7];
    const float* dbhh  = (const float*)d_in[8];
    const float* linW  = (const float*)d_in[9];
    const float* linb  = (const float*)d_in[10];
    float* out = (float*)d_out;

    const size_t BH = (size_t)BB * HH;
    float* ws   = (float*)d_ws;
    float* h_a  = ws;
    float* c_a  = ws + BH;
    float* h_b  = ws + 2 * BH;
    float* c_b  = ws + 3 * BH;
    float* hbuf = ws + 4 * BH;
    unsigned* bar = (unsigned*)(ws + 4 * BH + 2 * HH);

    // zero h0/c0 + hbuf + barrier counter (re-zeroed on every launch/replay)
    {
        const int n = (int)(2 * BH);
        zero_kernel<<<(n + 255) / 256, 256, 0, stream>>>(h_a, n);
        zero_kernel<<<(2 * HH + 1 + 255) / 256, 256, 0, stream>>>(hbuf, 2 * HH + 1);
    }

    // Encoder: 64 sequential fused WMMA steps, ping-pong state
    const float *hi = h_a, *ci = c_a;
    float *ho = h_b, *co = c_b;
    for (int t = 0; t < AA; ++t) {
        enc_step_kernel<<<128, 128, 0, stream>>>(
            input + (size_t)t * BB * DD, eWih, eWhh, ebih, ebhh,
            hi, ci, ho, co);
        const float* th = hi; hi = ho; ho = (float*)th;
        const float* tc = ci; ci = co; co = (float*)tc;
    }

    // Decoder: 64 cooperating WGPs, LDS-resident weights, TDM x prefetch
    dec_kernel<<<DEC_WGS, DEC_THR, 0, stream>>>(
        input, dWih, dWhh, dbih, dbhh, linW, linb, hi, ci, hbuf, bar, out);
}